// DAKPConvX_16157666968109
// MI455X (gfx1250) — compile-verified
//
#include <hip/hip_runtime.h>
#include <math.h>

// ---------------------------------------------------------------------------
// DAKPConvX fused kernel for MI455X (gfx1250, wave32).
//
// out[m,c] = sum_h infl[m,h] * s_feats[idx[m,h],c] * weights[nn[m,h],c]
//                 * sigmoid( leaky_relu(s_feats[m]@w1 + b1) @ w2 )[m, nn*16 + c/8]
//
// The two dense per-point GEMMs run on the CDNA5 matrix pipe with the
// full-precision V_WMMA_F32_16X16X4_F32 instruction (f32 A/B/C keeps us
// bit-close to the reference; the GEMM is only ~10 GFLOP so fp32-rate WMMA
// is already far off the critical path — the kernel is L2-gather bound).
// ---------------------------------------------------------------------------

typedef float v2f __attribute__((ext_vector_type(2)));
typedef float v8f __attribute__((ext_vector_type(8)));

#define CC     128          // channels
#define HH     32           // neighbors
#define KK     15           // kernel points
#define CPG    16           // channels-per-group = C / GROUPS
#define W2N    (KK * CPG)   // 240 = width of second GEMM
#define TILE_M 16           // query points per block
#define LDA    132          // padded LDS stride (132 % 64 = 4 -> no bank conflicts)

__global__ __launch_bounds__(128)
void dakp_fused_kernel(const float* __restrict__ q_pts,        // [M,3]
                       const float* __restrict__ s_pts,        // [N,3]
                       const float* __restrict__ s_feats,      // [N,C]
                       const int*   __restrict__ neighb_inds,  // [M,H]
                       const float* __restrict__ da_scale,     // [M]
                       const float* __restrict__ weights,      // [K,C]
                       const float* __restrict__ kernel_points,// [K,3]
                       const float* __restrict__ w1,           // [C,C]
                       const float* __restrict__ b1,           // [C]
                       const float* __restrict__ w2,           // [C, K*CPG]
                       float*       __restrict__ out,          // [M,C]
                       int M, int N)
{
    const int tid  = threadIdx.x;
    const int lane = tid & 31;
    const int wv   = tid >> 5;        // wave id 0..3
    const int half = lane >> 4;       // 0: lanes 0-15, 1: lanes 16-31
    const int l16  = lane & 15;
    const int m0   = blockIdx.x * TILE_M;

    __shared__ float As[TILE_M][LDA];     // s_feats tile (GEMM1 A operand)
    __shared__ float Hs[TILE_M][LDA];     // hidden layer (GEMM2 A operand)
    __shared__ float modS[TILE_M][W2N];   // sigmoid modulations [16][240]

    // ---- stage 16x128 s_feats tile into LDS (coalesced float4) -------------
    for (int i = tid; i < TILE_M * (CC / 4); i += blockDim.x) {
        const int r  = i >> 5;            // CC/4 == 32 float4 per row
        const int c4 = i & 31;
        int mr = m0 + r; if (mr >= M) mr = M - 1;
        const float4 v =
            *reinterpret_cast<const float4*>(s_feats + (size_t)mr * CC + c4 * 4);
        As[r][c4 * 4 + 0] = v.x;  As[r][c4 * 4 + 1] = v.y;
        As[r][c4 * 4 + 2] = v.z;  As[r][c4 * 4 + 3] = v.w;
    }
    __syncthreads();

    // ---- GEMM1: Hs = leaky_relu(As @ w1 + b1), 8 output tiles of 16 --------
    // V_WMMA_F32_16X16X4_F32 operand layout (ISA 7.12.2):
    //   A 16x4 : lane -> M=l16, VGPR0/1 hold K = half*2 + {0,1}
    //   B 4x16 : lane -> N=l16, VGPR0/1 hold K = half*2 + {0,1}
    //   C/D    : lane -> N=l16, VGPR r -> M = r + half*8
    for (int nt = wv; nt < CC / 16; nt += 4) {
        v8f acc = {0.f, 0.f, 0.f, 0.f, 0.f, 0.f, 0.f, 0.f};
        const int n = nt * 16 + l16;
        for (int kc = 0; kc < CC / 4; ++kc) {
            const int kb = kc * 4 + half * 2;
            v2f a, b;
            a.x = As[l16][kb + 0];
            a.y = As[l16][kb + 1];
            b.x = w1[(size_t)(kb + 0) * CC + n];
            b.y = w1[(size_t)(kb + 1) * CC + n];
            acc = __builtin_amdgcn_wmma_f32_16x16x4_f32(
                      false, a, false, b, (short)0, acc, false, false);
        }
        const float bias = b1[n];
#pragma unroll
        for (int r = 0; r < 8; ++r) {
            float v = acc[r] + bias;
            Hs[r + half * 8][n] = (v > 0.f) ? v : 0.1f * v;   // leaky_relu(0.1)
        }
    }
    __syncthreads();

    // ---- GEMM2: modS = sigmoid(Hs @ w2), 15 output tiles of 16 -------------
    for (int nt = wv; nt < W2N / 16; nt += 4) {
        v8f acc = {0.f, 0.f, 0.f, 0.f, 0.f, 0.f, 0.f, 0.f};
        const int n = nt * 16 + l16;
        for (int kc = 0; kc < CC / 4; ++kc) {
            const int kb = kc * 4 + half * 2;
            v2f a, b;
            a.x = Hs[l16][kb + 0];
            a.y = Hs[l16][kb + 1];
            b.x = w2[(size_t)(kb + 0) * W2N + n];
            b.y = w2[(size_t)(kb + 1) * W2N + n];
            acc = __builtin_amdgcn_wmma_f32_16x16x4_f32(
                      false, a, false, b, (short)0, acc, false, false);
        }
#pragma unroll
        for (int r = 0; r < 8; ++r) {
            modS[r + half * 8][n] = 1.0f / (1.0f + expf(-acc[r]));
        }
    }
    __syncthreads();

    // ---- phase 3: influence + gather-weighted reduction --------------------
    // 4 points per wave; one lane per neighbor for the geometry, then the
    // 32 lanes sweep the 128 channels (4 each, stride 32 -> coalesced).
    for (int pi = 0; pi < 4; ++pi) {
        const int p = wv * 4 + pi;
        const int m = m0 + p;
        if (m >= M) continue;                 // wave-uniform

        int   idx  = neighb_inds[(size_t)m * HH + lane];
        float infl = 0.f;
        int   nn   = 0;
        if ((unsigned)idx < (unsigned)N) {    // shadow neighbors -> infl 0
            const float qx = q_pts[(size_t)m * 3 + 0];
            const float qy = q_pts[(size_t)m * 3 + 1];
            const float qz = q_pts[(size_t)m * 3 + 2];
            const float dx = s_pts[(size_t)idx * 3 + 0] - qx;
            const float dy = s_pts[(size_t)idx * 3 + 1] - qy;
            const float dz = s_pts[(size_t)idx * 3 + 2] - qz;
            const float da = da_scale[m];
            float best = 3.4e38f;
#pragma unroll
            for (int k = 0; k < KK; ++k) {
                const float ex = dx - kernel_points[k * 3 + 0] * da;
                const float ey = dy - kernel_points[k * 3 + 1] * da;
                const float ez = dz - kernel_points[k * 3 + 2] * da;
                const float d2 = ex * ex + ey * ey + ez * ez;
                if (d2 < best) { best = d2; nn = k; }   // first-min like argmin
            }
            const float t = 1.0f - sqrtf(best);         // SIGMA = 1
            infl = (t > 0.f) ? t : 0.f;
        } else {
            idx = 0;                                    // dead, infl == 0
        }

        float acc0 = 0.f, acc1 = 0.f, acc2 = 0.f, acc3 = 0.f;
        for (int h = 0; h < HH; ++h) {
            const float fI  = __shfl(infl, h, 32);      // uniform after shfl
            const int   iI  = __shfl(idx,  h, 32);
            const int   nnh = __shfl(nn,   h, 32);
            if (fI > 0.f) {
                const float* fr = s_feats + (size_t)iI  * CC;
                const float* wr = weights + (size_t)nnh * CC;
                const float* mr = &modS[p][nnh * CPG];
                acc0 += fI * fr[lane +  0] * wr[lane +  0] * mr[(lane +  0) >> 3];
                acc1 += fI * fr[lane + 32] * wr[lane + 32] * mr[(lane + 32) >> 3];
                acc2 += fI * fr[lane + 64] * wr[lane + 64] * mr[(lane + 64) >> 3];
                acc3 += fI * fr[lane + 96] * wr[lane + 96] * mr[(lane + 96) >> 3];
            }
        }
        float* op = out + (size_t)m * CC;
        op[lane +  0] = acc0;
        op[lane + 32] = acc1;
        op[lane + 64] = acc2;
        op[lane + 96] = acc3;
    }
}

extern "C" void kernel_launch(void* const* d_in, const int* in_sizes, int n_in,
                              void* d_out, int out_size, void* d_ws, size_t ws_size,
                              hipStream_t stream) {
    (void)n_in; (void)out_size; (void)d_ws; (void)ws_size;

    const float* q_pts         = (const float*)d_in[0];
    const float* s_pts         = (const float*)d_in[1];
    const float* s_feats       = (const float*)d_in[2];
    const int*   neighb_inds   = (const int*)  d_in[3];
    const float* da_scale      = (const float*)d_in[4];
    const float* weights       = (const float*)d_in[5];
    const float* kernel_points = (const float*)d_in[6];
    const float* w1            = (const float*)d_in[7];
    const float* b1            = (const float*)d_in[8];
    const float* w2            = (const float*)d_in[9];
    float*       out           = (float*)d_out;

    const int M = in_sizes[4];        // da_scale has M elements
    const int N = in_sizes[1] / 3;    // s_pts is [N,3]

    const int grid = (M + TILE_M - 1) / TILE_M;
    dakp_fused_kernel<<<grid, 128, 0, stream>>>(
        q_pts, s_pts, s_feats, neighb_inds, da_scale,
        weights, kernel_points, w1, b1, w2, out, M, N);
}